// LocalLinear_20255065768535
// MI455X (gfx1250) — compile-verified
//
#include <hip/hip_runtime.h>

typedef __attribute__((ext_vector_type(2))) float v2f;
typedef __attribute__((ext_vector_type(8))) float v8f;

#define N_ROWS   1024
#define P_DIM    20000
#define K_TAPS   8
#define P_BLOCK  128            // 8 waves * 16 columns
#define XCOLS    (P_BLOCK + 8)  // 128 + (K-1) halo, rounded to 136
#define XSTRIDE  137            // +1 pad: 9*row mod 64 distinct -> conflict-free column reads

__global__ __launch_bounds__(256)
void local_linear_wmma(const float* __restrict__ X,
                       const float* __restrict__ W,
                       const float* __restrict__ Bias,
                       float* __restrict__ Out)
{
    __shared__ float Xs[16 * XSTRIDE];
    __shared__ float Ws[P_BLOCK * K_TAPS];

    const int tid = threadIdx.x;
    const int px0 = blockIdx.x * P_BLOCK;   // first p column staged by this block
    const int n0  = blockIdx.y * 16;        // first sample row of this block

    // ---- Stage X tile (16 rows x 136 cols), coalesced along columns; zero-pad past P ----
    for (int i = tid; i < 16 * XCOLS; i += 256) {
        int r  = i / XCOLS;
        int c  = i - r * XCOLS;
        int gc = px0 + c;
        float v = 0.0f;
        if (gc < P_DIM) v = X[(size_t)(n0 + r) * P_DIM + gc];
        Xs[r * XSTRIDE + c] = v;
    }
    // ---- Stage weights: 128 positions x 8 taps, coalesced ----
    for (int i = tid; i < P_BLOCK * K_TAPS; i += 256) {
        int gp = px0 + (i >> 3);
        Ws[i] = (gp < P_DIM) ? W[(size_t)gp * K_TAPS + (i & 7)] : 0.0f;
    }
    __syncthreads();

    const int wave = tid >> 5;
    const int lane = tid & 31;
    const int lrow = lane & 15;     // = M (A rows) = N (B/C/D cols) fragment index
    const int lhi  = lane >> 4;     // lane-half selector
    const int pw0  = px0 + wave * 16;
    if (pw0 >= P_DIM) return;       // wave-uniform: EXEC stays all-1 for active waves

    v8f acc = {};
    const float* xrow = &Xs[lrow * XSTRIDE + wave * 16];
    const float* wrow = &Ws[(wave * 16 + lrow) << 3];

    #pragma unroll
    for (int j = 0; j < 6; ++j) {
        // A 16x4 fragment: lane holds M=lrow, K = 2*lhi + {0,1}  (q offset within window)
        const int q0 = 4 * j + 2 * lhi;
        const int q1 = q0 + 1;
        v2f a, b;
        a.x = xrow[q0];
        a.y = xrow[q1];
        // B 4x16 banded fragment: B[K,N] = W[pw0+N, K-N] if 0<=K-N<8 else 0
        const int k0 = q0 - lrow;
        const int k1 = q1 - lrow;
        float w0 = wrow[(unsigned)k0 & 7u];   // clamped in-bounds read, masked below
        float w1 = wrow[(unsigned)k1 & 7u];
        b.x = ((unsigned)k0 < 8u) ? w0 : 0.0f;
        b.y = ((unsigned)k1 < 8u) ? w1 : 0.0f;
        acc = __builtin_amdgcn_wmma_f32_16x16x4_f32(
            /*neg_a=*/false, a, /*neg_b=*/false, b,
            /*c_mod=*/(short)0, acc, /*reuse_a=*/false, /*reuse_b=*/false);
    }

    // ---- Store D: VGPR v holds (M = v + 8*lhi, N = lrow); add per-column bias ----
    const float bias = Bias[pw0 + lrow];
    size_t base = (size_t)(n0 + 8 * lhi) * P_DIM + (size_t)(pw0 + lrow);
    #pragma unroll
    for (int v = 0; v < 8; ++v) {
        Out[base + (size_t)v * P_DIM] = acc[v] + bias;
    }
}

extern "C" void kernel_launch(void* const* d_in, const int* in_sizes, int n_in,
                              void* d_out, int out_size, void* d_ws, size_t ws_size,
                              hipStream_t stream) {
    (void)in_sizes; (void)n_in; (void)d_ws; (void)ws_size; (void)out_size;
    const float* X    = (const float*)d_in[0];   // (1024, 20000)
    const float* W    = (const float*)d_in[1];   // (20000, 8, 1)
    const float* Bias = (const float*)d_in[2];   // (20000, 1)
    float* Out        = (float*)d_out;           // (1024, 20000)

    dim3 grid((P_DIM + P_BLOCK - 1) / P_BLOCK, N_ROWS / 16);
    local_linear_wmma<<<grid, dim3(256), 0, stream>>>(X, W, Bias, Out);
}